// PCircuit_60060822667786
// MI455X (gfx1250) — compile-verified
//
#include <hip/hip_runtime.h>
#include <math.h>

#define NN 4096  // number of p-bits (matches reference N)

typedef __attribute__((ext_vector_type(2))) float v2f;
typedef __attribute__((ext_vector_type(8))) float v8f;

// ---------------------------------------------------------------------------
// Kernel 1: F = J @ m0 + h  via V_WMMA_F32_16X16X4_F32 (pure fp32 WMMA).
// One wave per 16-row tile (256 tiles). B is the m0 slice broadcast across all
// 16 columns, so every column of D holds the same dot product -> N-lane layout
// is irrelevant; K layout follows the documented A-matrix pattern
// (VGPR0: K=0 lanes0-15 / K=2 lanes16-31; VGPR1: K=1 / K=3).
// Side effect: streams all 64 MB of J once, warming the 192 MB L2 for kernel 2.
// ---------------------------------------------------------------------------
__global__ __launch_bounds__(256) void pbit_init_field(
    const float* __restrict__ J, const float* __restrict__ h,
    const float* __restrict__ m0, float* __restrict__ F) {
  const int lane = threadIdx.x & 31;
  const int tile = blockIdx.x * 8 + (threadIdx.x >> 5);   // 32 blocks * 8 waves
  const int rowInTile = (lane < 16) ? lane : (lane - 16);
  const int row = tile * 16 + rowInTile;
  const int kOff = (lane < 16) ? 0 : 2;                   // K placement per lane half
  const float* __restrict__ Jrow = J + (size_t)row * NN;

  v8f c0 = {}, c1 = {}, c2 = {}, c3 = {};                 // 4 chains to hide WMMA latency
  for (int k = 0; k < NN; k += 16) {
    v2f a0, a1, a2, a3, b0, b1, b2, b3;
    a0.x = Jrow[k + 0 + kOff];  a0.y = Jrow[k + 1 + kOff];
    a1.x = Jrow[k + 4 + kOff];  a1.y = Jrow[k + 5 + kOff];
    a2.x = Jrow[k + 8 + kOff];  a2.y = Jrow[k + 9 + kOff];
    a3.x = Jrow[k + 12 + kOff]; a3.y = Jrow[k + 13 + kOff];
    b0.x = m0[k + 0 + kOff];    b0.y = m0[k + 1 + kOff];
    b1.x = m0[k + 4 + kOff];    b1.y = m0[k + 5 + kOff];
    b2.x = m0[k + 8 + kOff];    b2.y = m0[k + 9 + kOff];
    b3.x = m0[k + 12 + kOff];   b3.y = m0[k + 13 + kOff];
    c0 = __builtin_amdgcn_wmma_f32_16x16x4_f32(false, a0, false, b0, (short)0, c0, false, false);
    c1 = __builtin_amdgcn_wmma_f32_16x16x4_f32(false, a1, false, b1, (short)0, c1, false, false);
    c2 = __builtin_amdgcn_wmma_f32_16x16x4_f32(false, a2, false, b2, (short)0, c2, false, false);
    c3 = __builtin_amdgcn_wmma_f32_16x16x4_f32(false, a3, false, b3, (short)0, c3, false, false);
  }
  v8f c = c0 + c1 + c2 + c3;

  // D layout (documented): VGPR r, lane 0 -> (M=r, N=0); lane 16 -> (M=8+r, N=0)
  if (lane == 0) {
#pragma unroll
    for (int r = 0; r < 8; ++r) F[tile * 16 + r] = c[r] + h[tile * 16 + r];
  } else if (lane == 16) {
#pragma unroll
    for (int r = 0; r < 8; ++r) F[tile * 16 + 8 + r] = c[r] + h[tile * 16 + 8 + r];
  }
}

// ---------------------------------------------------------------------------
// Kernel 2: 16384 sequential Glauber updates. Single persistent workgroup of
// 1024 threads (32 wave32s on one WGP). Each thread owns 4 consecutive F
// entries in registers; spins + the broadcast delta live in LDS. One barrier
// per step, with the (t+1) owner computing its update *after* applying step
// t's rank-1 correction, using a parity-double-buffered delta slot. J rows
// are L2-resident (warmed by kernel 1); next row is prefetched each step.
// ---------------------------------------------------------------------------
__global__ __launch_bounds__(1024) void pbit_glauber(
    const float* __restrict__ J, const float* __restrict__ m0,
    const int* __restrict__ idx, const float* __restrict__ u,
    const float* __restrict__ F_in, float* __restrict__ out, int steps) {
  __shared__ float m_s[NN];
  __shared__ float deltaSlot[2];
  const int tid = threadIdx.x;

  for (int j = tid; j < NN; j += 1024) m_s[j] = m0[j];
  float4 f = reinterpret_cast<const float4*>(F_in)[tid];  // F[4*tid .. 4*tid+3]
  __syncthreads();  // m_s fully initialized before any owner reads it

  // Seed step 0.
  {
    const int i0 = idx[0];
    if (tid == (i0 >> 2)) {
      const int q = i0 & 3;
      const float Ii = (q == 0) ? f.x : (q == 1) ? f.y : (q == 2) ? f.z : f.w;
      const float p = tanhf(Ii);                  // BETA = 1
      const float r = 2.0f * u[0] - 1.0f;
      const float s = (p - r >= 0.0f) ? 1.0f : -1.0f;  // sign with 0 -> +1
      deltaSlot[0] = s - m_s[i0];
      m_s[i0] = s;
    }
  }
  __syncthreads();

  for (int t = 0; t < steps; ++t) {
    const float delta = deltaSlot[t & 1];
    const int i = idx[t];
    const bool hasNext = (t + 1 < steps);
    const int inext = hasNext ? idx[t + 1] : 0;
    if (hasNext)  // warm next step's J row across the barrier (global_prefetch_b8)
      __builtin_prefetch(J + (size_t)inext * NN + 4 * tid, 0, 3);

    if (delta != 0.0f) {  // uniform branch across the block
      const float4 j4 =
          reinterpret_cast<const float4*>(J + (size_t)i * NN)[tid];
      f.x += delta * j4.x;
      f.y += delta * j4.y;
      f.z += delta * j4.z;
      f.w += delta * j4.w;
    }

    if (hasNext && tid == (inext >> 2)) {  // owner computes step t+1 post-update
      const int q = inext & 3;
      const float Ii = (q == 0) ? f.x : (q == 1) ? f.y : (q == 2) ? f.z : f.w;
      const float p = tanhf(Ii);
      const float r = 2.0f * u[t + 1] - 1.0f;
      const float s = (p - r >= 0.0f) ? 1.0f : -1.0f;
      deltaSlot[(t + 1) & 1] = s - m_s[inext];
      m_s[inext] = s;
    }
    __syncthreads();
  }

  for (int j = tid; j < NN; j += 1024) out[j] = m_s[j];
}

// ---------------------------------------------------------------------------
// Host launcher. Inputs (setup_inputs order): J[N*N] f32, h[N] f32, m0[N] f32,
// idx[STEPS] i32, u[STEPS] f32. Output: m[N] f32. Workspace: F[N] f32 (16 KB).
// ---------------------------------------------------------------------------
extern "C" void kernel_launch(void* const* d_in, const int* in_sizes, int n_in,
                              void* d_out, int out_size, void* d_ws,
                              size_t ws_size, hipStream_t stream) {
  const float* J = (const float*)d_in[0];
  const float* h = (const float*)d_in[1];
  const float* m0 = (const float*)d_in[2];
  const int* idx = (const int*)d_in[3];
  const float* u = (const float*)d_in[4];
  float* out = (float*)d_out;
  float* F = (float*)d_ws;  // N floats of scratch
  const int steps = in_sizes[3];

  pbit_init_field<<<dim3(32), dim3(256), 0, stream>>>(J, h, m0, F);
  pbit_glauber<<<dim3(1), dim3(1024), 0, stream>>>(J, m0, idx, u, F, out,
                                                   steps);
}